// GNO_76733885710904
// MI455X (gfx1250) — compile-verified
//
#include <hip/hip_runtime.h>

typedef __attribute__((ext_vector_type(2))) float v2f;
typedef __attribute__((ext_vector_type(8))) float v8f;

__device__ __forceinline__ float gelu_exact(float x) {
    return 0.5f * x * (1.0f + erff(x * 0.70710678118654752440f));
}

// D = A(16xK) * B(Kx16) + C via V_WMMA_F32_16X16X4_F32 (exact f32)
__device__ __forceinline__ v8f wmma4(v2f a, v2f b, v8f c) {
    return __builtin_amdgcn_wmma_f32_16x16x4_f32(false, a, false, b, (short)0, c, false, false);
}

// A-frag (16x4 slice at k-step ks) from row-major X[16][ld]:
// ISA 7.12.2: lane<16 holds K=0,1 ; lane>=16 holds K=2,3 ; M = lane&15
__device__ __forceinline__ v2f load_a(const float* __restrict__ X, int ld, int ks, int lane) {
    int M = lane & 15;
    int k = ks * 4 + ((lane >> 4) << 1);
    const float* p = X + (size_t)M * ld + k;
    return v2f{p[0], p[1]};
}
// B-frag (4x16 slice) from row-major W[K][ldw], column block nc..nc+15 (mirror layout)
__device__ __forceinline__ v2f load_b(const float* __restrict__ W, int ld, int ks, int nc, int lane) {
    int c = nc + (lane & 15);
    int k = ks * 4 + ((lane >> 4) << 1);
    return v2f{W[(size_t)k * ld + c], W[(size_t)(k + 1) * ld + c]};
}
// C/D frag -> row-major O[16][ld], column tile nt. M = r + 8*(lane>=16), N = nt*16 + (lane&15)
__device__ __forceinline__ void store_cd(float* __restrict__ O, int ld, int nt, int lane, v8f c) {
    int M0 = (lane >> 4) << 3;
    int col = nt * 16 + (lane & 15);
#pragma unroll
    for (int r = 0; r < 8; ++r) O[(size_t)(M0 + r) * ld + col] = c[r];
}

// ---------- projection layer 1: t0 = GELU(cat(x,grid) @ W1 + b1), K=12 ----------
__global__ void k_proj1(const float* __restrict__ x, const float* __restrict__ grid,
                        const float* __restrict__ W1, const float* __restrict__ b1,
                        float* __restrict__ t0, int n) {
    int lane = threadIdx.x & 31;
    int tile = blockIdx.x * (blockDim.x >> 5) + (threadIdx.x >> 5);
    int node0 = tile * 16;
    if (node0 >= n) return;
    int M = lane & 15;
    v2f af[3];
#pragma unroll
    for (int ks = 0; ks < 3; ++ks) {
        int k = ks * 4 + ((lane >> 4) << 1);
        float a0 = (k < 10)     ? x[(size_t)(node0 + M) * 10 + k]
                                : grid[(size_t)(node0 + M) * 2 + (k - 10)];
        float a1 = (k + 1 < 10) ? x[(size_t)(node0 + M) * 10 + k + 1]
                                : grid[(size_t)(node0 + M) * 2 + (k + 1 - 10)];
        af[ks] = v2f{a0, a1};
    }
#pragma unroll
    for (int nt = 0; nt < 8; ++nt) {
        v8f acc = {};
#pragma unroll
        for (int ks = 0; ks < 3; ++ks)
            acc = wmma4(af[ks], load_b(W1, 128, ks, nt * 16, lane), acc);
        float bv = b1[nt * 16 + (lane & 15)];
#pragma unroll
        for (int r = 0; r < 8; ++r) acc[r] = gelu_exact(acc[r] + bv);
        store_cd(t0 + (size_t)node0 * 128, 128, nt, lane, acc);
    }
}

// ---------- generic row-tile GEMM: O = act(X @ W + bias) ----------
template <int KSTEPS, int NTILES, bool DO_GELU>
__global__ void k_gemm(const float* __restrict__ X, int ldx,
                       const float* __restrict__ W, int ldw,
                       const float* __restrict__ bias,
                       float* __restrict__ O, int ldo, int n) {
    int lane = threadIdx.x & 31;
    int tile = blockIdx.x * (blockDim.x >> 5) + (threadIdx.x >> 5);
    int node0 = tile * 16;
    if (node0 >= n) return;
    v2f af[KSTEPS];
#pragma unroll
    for (int ks = 0; ks < KSTEPS; ++ks)
        af[ks] = load_a(X + (size_t)node0 * ldx, ldx, ks, lane);
#pragma unroll
    for (int nt = 0; nt < NTILES; ++nt) {
        v8f acc = {};
#pragma unroll
        for (int ks = 0; ks < KSTEPS; ++ks)
            acc = wmma4(af[ks], load_b(W, ldw, ks, nt * 16, lane), acc);
        float bv = bias[nt * 16 + (lane & 15)];
#pragma unroll
        for (int r = 0; r < 8; ++r) {
            float v = acc[r] + bv;
            acc[r] = DO_GELU ? gelu_exact(v) : v;
        }
        store_cd(O + (size_t)node0 * ldo, ldo, nt, lane, acc);
    }
}

// ---------- node-static edge-MLP parts + Ww: A = h@Wk[5:69]+grid@Wk[1:3]+bk ;
//            B = h@Wk[69:133]+grid@Wk[3:5] ; hw = h@Ww + bw ----------
__global__ void k_nodeprep(const float* __restrict__ h, const float* __restrict__ grid,
                           const float* __restrict__ Wk, const float* __restrict__ bk,
                           const float* __restrict__ Ww, const float* __restrict__ bw,
                           float* __restrict__ A, float* __restrict__ Bn,
                           float* __restrict__ hw, int n) {
    int lane = threadIdx.x & 31;
    int tile = blockIdx.x * (blockDim.x >> 5) + (threadIdx.x >> 5);
    int node0 = tile * 16;
    if (node0 >= n) return;
    int M = lane & 15;
    int khalf = lane >> 4;
    v2f af[16];
#pragma unroll
    for (int ks = 0; ks < 16; ++ks)
        af[ks] = load_a(h + (size_t)node0 * 64, 64, ks, lane);
    // rank-2 grid term as one extra K=4 WMMA (K=2,3 zero-padded)
    v2f ag = khalf ? v2f{0.f, 0.f}
                   : v2f{grid[(size_t)(node0 + M) * 2], grid[(size_t)(node0 + M) * 2 + 1]};
#pragma unroll
    for (int nt = 0; nt < 4; ++nt) {
        int c = nt * 16 + (lane & 15);
        v8f aA = {}, aB = {}, aW = {};
#pragma unroll
        for (int ks = 0; ks < 16; ++ks) {
            v2f a = af[ks];
            aA = wmma4(a, load_b(Wk + 5 * 64, 64, ks, nt * 16, lane), aA);
            aB = wmma4(a, load_b(Wk + 69 * 64, 64, ks, nt * 16, lane), aB);
            aW = wmma4(a, load_b(Ww, 64, ks, nt * 16, lane), aW);
        }
        v2f bgi = khalf ? v2f{0.f, 0.f} : v2f{Wk[64 + c], Wk[128 + c]};   // rows 1,2 (g_i)
        v2f bgj = khalf ? v2f{0.f, 0.f} : v2f{Wk[192 + c], Wk[256 + c]};  // rows 3,4 (g_j)
        aA = wmma4(ag, bgi, aA);
        aB = wmma4(ag, bgj, aB);
        float bkv = bk[c], bwv = bw[c];
#pragma unroll
        for (int r = 0; r < 8; ++r) { aA[r] += bkv; aW[r] += bwv; }
        store_cd(A + (size_t)node0 * 64, 64, nt, lane, aA);
        store_cd(Bn + (size_t)node0 * 64, 64, nt, lane, aB);
        store_cd(hw + (size_t)node0 * 64, 64, nt, lane, aW);
    }
}

// ---------- per-edge pass: gather B[src] (L2-resident), scatter-add into Bsum[dst],
//            plus dist/deg scalars. 16 lanes per edge, float4 per lane ----------
__global__ void k_edge(const int* __restrict__ ei, const float* __restrict__ grid,
                       const float* __restrict__ Bn,
                       float* __restrict__ Bsum, float* __restrict__ dsum,
                       float* __restrict__ deg, int nE) {
    int t = blockIdx.x * blockDim.x + threadIdx.x;
    int e = t >> 4;
    int sub = t & 15;
    if (e >= nE) return;
    int src = ei[e];       // edge_index row 0 = j (source)
    int dst = ei[nE + e];  // edge_index row 1 = i (target)
    float4 bv = ((const float4*)(Bn + (size_t)src * 64))[sub];
    float* o = Bsum + (size_t)dst * 64 + sub * 4;
    atomicAdd(o + 0, bv.x);
    atomicAdd(o + 1, bv.y);
    atomicAdd(o + 2, bv.z);
    atomicAdd(o + 3, bv.w);
    if (sub == 0) {
        float gx = grid[(size_t)dst * 2]     - grid[(size_t)src * 2];
        float gy = grid[(size_t)dst * 2 + 1] - grid[(size_t)src * 2 + 1];
        atomicAdd(dsum + dst, sqrtf(gx * gx + gy * gy));
        atomicAdd(deg + dst, 1.0f);
    }
}

// ---------- u = GELU(hw + deg*A + Bsum + dsum*Wdist), written in place over Bsum ----------
__global__ void k_update(const float* __restrict__ hw, const float* __restrict__ A,
                         const float* __restrict__ Wk, const float* __restrict__ dsum,
                         const float* __restrict__ deg, float* __restrict__ Bsum_u, int n) {
    int t = blockIdx.x * blockDim.x + threadIdx.x;
    int m = t >> 4;
    int sub = t & 15;
    if (m >= n) return;
    int c = sub * 4;
    float dg = deg[m], ds = dsum[m];
    float4 a  = *(const float4*)(A + (size_t)m * 64 + c);
    float4 w  = *(const float4*)(hw + (size_t)m * 64 + c);
    float4 bs = *(const float4*)(Bsum_u + (size_t)m * 64 + c);
    float4 wd = *(const float4*)(Wk + c);  // Wk row 0 = dist weights
    float4 u;
    u.x = gelu_exact(w.x + dg * a.x + bs.x + ds * wd.x);
    u.y = gelu_exact(w.y + dg * a.y + bs.y + ds * wd.y);
    u.z = gelu_exact(w.z + dg * a.z + bs.z + ds * wd.z);
    u.w = gelu_exact(w.w + dg * a.w + bs.w + ds * wd.w);
    *(float4*)(Bsum_u + (size_t)m * 64 + c) = u;
}

// ---------- final decode dot: out[m] = t1[m,:] . Wd2 + bd2 (wave per node) ----------
__global__ void k_dec2(const float* __restrict__ t1, const float* __restrict__ Wd2,
                       const float* __restrict__ bd2, float* __restrict__ out, int n) {
    int lane = threadIdx.x & 31;
    int node = blockIdx.x * (blockDim.x >> 5) + (threadIdx.x >> 5);
    if (node >= n) return;
    float4 tv = *(const float4*)(t1 + (size_t)node * 128 + lane * 4);
    float4 wv = *(const float4*)(Wd2 + lane * 4);
    float s = tv.x * wv.x + tv.y * wv.y + tv.z * wv.z + tv.w * wv.w;
#pragma unroll
    for (int msk = 16; msk > 0; msk >>= 1) s += __shfl_xor(s, msk, 32);
    if (lane == 0) out[node] = s + bd2[0];
}

extern "C" void kernel_launch(void* const* d_in, const int* in_sizes, int n_in,
                              void* d_out, int out_size, void* d_ws, size_t ws_size,
                              hipStream_t stream) {
    (void)n_in; (void)out_size; (void)ws_size;
    const float* x    = (const float*)d_in[0];
    const float* grid = (const float*)d_in[1];
    const int*   ei   = (const int*)d_in[2];
    const float* W1   = (const float*)d_in[3];
    const float* b1   = (const float*)d_in[4];
    const float* W2   = (const float*)d_in[5];
    const float* b2   = (const float*)d_in[6];
    const float* Wk   = (const float*)d_in[7];
    const float* bk   = (const float*)d_in[8];
    const float* Ww   = (const float*)d_in[9];
    const float* bw   = (const float*)d_in[10];
    const float* Wd1  = (const float*)d_in[11];
    const float* bd1  = (const float*)d_in[12];
    const float* Wd2  = (const float*)d_in[13];
    const float* bd2  = (const float*)d_in[14];
    float* out = (float*)d_out;

    int n  = in_sizes[0] / 10;  // 65536
    int nE = in_sizes[2] / 2;   // 1048576

    float* ws = (float*)d_ws;
    size_t off = 0;
    float* t0   = ws + off; off += (size_t)n * 128;  // also reused as t1
    float* h    = ws + off; off += (size_t)n * 64;
    float* A    = ws + off; off += (size_t)n * 64;
    float* Bn   = ws + off; off += (size_t)n * 64;
    float* hw   = ws + off; off += (size_t)n * 64;
    float* Bsum = ws + off; off += (size_t)n * 64;   // also reused as u
    float* dsum = ws + off; off += (size_t)n;
    float* deg  = ws + off; off += (size_t)n;

    int tiles = n / 16;
    dim3 blk(128);            // 4 wave32s, one 16-node tile each
    dim3 grd(tiles / 4);

    // zero Bsum + dsum + deg (contiguous n*66 floats)
    hipMemsetAsync(Bsum, 0, (size_t)n * 66 * sizeof(float), stream);

    k_proj1<<<grd, blk, 0, stream>>>(x, grid, W1, b1, t0, n);
    k_gemm<32, 4, false><<<grd, blk, 0, stream>>>(t0, 128, W2, 64, b2, h, 64, n);
    k_nodeprep<<<grd, blk, 0, stream>>>(h, grid, Wk, bk, Ww, bw, A, Bn, hw, n);
    k_edge<<<dim3((unsigned)((nE * 16 + 255) / 256)), dim3(256), 0, stream>>>(
        ei, grid, Bn, Bsum, dsum, deg, nE);
    k_update<<<dim3((unsigned)((n * 16 + 255) / 256)), dim3(256), 0, stream>>>(
        hw, A, Wk, dsum, deg, Bsum, n);
    k_gemm<16, 8, true><<<grd, blk, 0, stream>>>(Bsum, 64, Wd1, 128, bd1, t0, 128, n);
    k_dec2<<<dim3((unsigned)(n / 8)), dim3(256), 0, stream>>>(t0, Wd2, bd2, out, n);
}